// HCGPool_85899345949
// MI455X (gfx1250) — compile-verified
//
#include <hip/hip_runtime.h>
#include <hip/hip_bf16.h>

// ---- problem constants (match reference) ----
#define Nn   131072
#define Bg   256
#define PER  512
#define Kc   32
#define Cc   128
#define Ee   2097152
#define EPSV 1e-10f

typedef float v2f __attribute__((ext_vector_type(2)));
typedef float v8f __attribute__((ext_vector_type(8)));

static __device__ __forceinline__ v8f wmma_f32(v2f a, v2f b, v8f c) {
  // (neg_a, A, neg_b, B, c_mod, C, reuse_a, reuse_b)
  return __builtin_amdgcn_wmma_f32_16x16x4_f32(false, a, false, b, (short)0, c,
                                               false, false);
}

// ---------------------------------------------------------------------------
// Kernel 0: zero the t = s^T A accumulator (16 MB, L2-resident)
// ---------------------------------------------------------------------------
__global__ void k_zero(float4* __restrict__ p, int n4) {
  int i = blockIdx.x * blockDim.x + threadIdx.x;
  if (i < n4) p[i] = make_float4(0.f, 0.f, 0.f, 0.f);
}

// ---------------------------------------------------------------------------
// Kernel 1: logits = xW + b via V_WMMA_F32_16X16X4_F32, then softmax -> s
// 256 threads = 8 waves; each wave owns a 16x32 logits tile.
// ---------------------------------------------------------------------------
__global__ void k_logits_softmax(const float* __restrict__ x,
                                 const float* __restrict__ W,
                                 const float* __restrict__ bias,
                                 float* __restrict__ s_out) {
  __shared__ float Wl[Cc * Kc];       // 16 KB: whole weight matrix
  __shared__ float tile[8][16 * Kc];  // 16 KB: per-wave logits staging

  const int tid = threadIdx.x;
  for (int i = tid; i < Cc * Kc; i += 256) Wl[i] = W[i];
  __syncthreads();

  const int wave = tid >> 5, lane = tid & 31;
  const int h = lane >> 4, lm = lane & 15;
  const int row_base = blockIdx.x * 128 + wave * 16;

  v8f c0 = {}, c1 = {};
  for (int k = 0; k < Cc; k += 4) {
    const int ko = k + 2 * h;  // lanes 0-15 -> K0/K1, lanes 16-31 -> K2/K3
    v2f a = *(const v2f*)(x + (size_t)(row_base + lm) * Cc + ko);
    v2f b0, b1;
    b0.x = Wl[ko * Kc + lm];        b0.y = Wl[(ko + 1) * Kc + lm];
    b1.x = Wl[ko * Kc + lm + 16];   b1.y = Wl[(ko + 1) * Kc + lm + 16];
    c0 = wmma_f32(a, b0, c0);       // clusters 0..15
    c1 = wmma_f32(a, b1, c1);       // clusters 16..31
  }

  const float bi0 = bias[lm], bi1 = bias[lm + 16];
#pragma unroll
  for (int v = 0; v < 8; ++v) {
    const int m = h * 8 + v;        // C/D layout: VGPR v -> row v (+8 hi half)
    tile[wave][m * Kc + lm]      = c0[v] + bi0;
    tile[wave][m * Kc + lm + 16] = c1[v] + bi1;
  }
  __syncthreads();

  if (lane < 16) {                  // one lane per row: softmax over K=32
    float vals[Kc];
    float mx = -3.4e38f;
#pragma unroll
    for (int j = 0; j < Kc; ++j) {
      vals[j] = tile[wave][lane * Kc + j];
      mx = fmaxf(mx, vals[j]);
    }
    float sum = 0.f;
#pragma unroll
    for (int j = 0; j < Kc; ++j) { vals[j] = __expf(vals[j] - mx); sum += vals[j]; }
    const float inv = 1.f / sum;
    float* o = s_out + (size_t)(row_base + lane) * Kc;
#pragma unroll
    for (int j = 0; j < Kc; ++j) o[j] = vals[j] * inv;
  }
}

// ---------------------------------------------------------------------------
// Kernel 2: fused edge scatter: t[g,k,dst%PER] += w_e * clamp(s[src,k])
// One wave per edge, lane = cluster k. t (16 MB) stays in the 192 MB L2.
// ---------------------------------------------------------------------------
__global__ void k_scatter_t(const int* __restrict__ src,
                            const int* __restrict__ dst,
                            const float* __restrict__ ea,
                            const float* __restrict__ s,
                            float* __restrict__ t) {
  const int e  = (blockIdx.x * 256 + threadIdx.x) >> 5;
  const int kk = threadIdx.x & 31;
  const int sn = src[e];
  const int g  = sn >> 9;                 // /PER
  const int m  = dst[e] & (PER - 1);      // %PER
  const float w  = ea[e];
  const float sv = fmaxf(s[(size_t)sn * Kc + kk], EPSV);
  atomicAdd(t + ((size_t)(g * Kc + kk) << 9) + m, w * sv);
}

// ---------------------------------------------------------------------------
// Kernel 3a: x_super[g] = clamp(s_g)^T @ x_g  (32x512 @ 512x128), WMMA f32.
// One block per graph; wave w owns output columns [16w, 16w+16).
// ---------------------------------------------------------------------------
__global__ void k_xsuper(const float* __restrict__ x,
                         const float* __restrict__ s,
                         float* __restrict__ xs) {
  const int g = blockIdx.x;
  const int wave = threadIdx.x >> 5, lane = threadIdx.x & 31;
  const int h = lane >> 4, lm = lane & 15;
  const int cb = wave * 16;
  const size_t gP = (size_t)g * PER;

  v8f cA = {}, cB = {};
  for (int k = 0; k < PER; k += 4) {
    const int ko = k + 2 * h;
    const float* s0 = s + (gP + ko) * Kc;
    const float* s1 = s + (gP + ko + 1) * Kc;
    v2f a0, a1, b;
    a0.x = fmaxf(s0[lm], EPSV);        a0.y = fmaxf(s1[lm], EPSV);
    a1.x = fmaxf(s0[lm + 16], EPSV);   a1.y = fmaxf(s1[lm + 16], EPSV);
    b.x = x[(gP + ko) * Cc + cb + lm];
    b.y = x[(gP + ko + 1) * Cc + cb + lm];
    cA = wmma_f32(a0, b, cA);          // clusters 0..15
    cB = wmma_f32(a1, b, cB);          // clusters 16..31
  }
#pragma unroll
  for (int v = 0; v < 8; ++v) {
    const int M = h * 8 + v;
    xs[((size_t)g * Kc + M)      * Cc + cb + lm] = cA[v];
    xs[((size_t)g * Kc + M + 16) * Cc + cb + lm] = cB[v];
  }
}

// ---------------------------------------------------------------------------
// Kernel 3b: adj_raw[g] = t[g] @ clamp(s_g)  (32x512 @ 512x32), WMMA f32.
// One wave per graph, 4 accumulator tiles.
// ---------------------------------------------------------------------------
__global__ void k_adjsuper(const float* __restrict__ t,
                           const float* __restrict__ s,
                           float* __restrict__ araw) {
  const int g = (blockIdx.x * 256 + threadIdx.x) >> 5;
  const int lane = threadIdx.x & 31;
  const int h = lane >> 4, lm = lane & 15;
  const size_t gP = (size_t)g * PER;

  v8f c00 = {}, c01 = {}, c10 = {}, c11 = {};
  for (int k = 0; k < PER; k += 4) {
    const int ko = k + 2 * h;
    v2f a0 = *(const v2f*)(t + ((size_t)(g * Kc) + lm)      * PER + ko);
    v2f a1 = *(const v2f*)(t + ((size_t)(g * Kc) + lm + 16) * PER + ko);
    const float* s0 = s + (gP + ko) * Kc;
    const float* s1 = s + (gP + ko + 1) * Kc;
    v2f b0, b1;
    b0.x = fmaxf(s0[lm], EPSV);        b0.y = fmaxf(s1[lm], EPSV);
    b1.x = fmaxf(s0[lm + 16], EPSV);   b1.y = fmaxf(s1[lm + 16], EPSV);
    c00 = wmma_f32(a0, b0, c00);
    c01 = wmma_f32(a0, b1, c01);
    c10 = wmma_f32(a1, b0, c10);
    c11 = wmma_f32(a1, b1, c11);
  }
  float* o = araw + (size_t)g * Kc * Kc;
#pragma unroll
  for (int v = 0; v < 8; ++v) {
    const int M = h * 8 + v;
    o[M * Kc + lm]             = c00[v];
    o[M * Kc + lm + 16]        = c01[v];
    o[(M + 16) * Kc + lm]      = c10[v];
    o[(M + 16) * Kc + lm + 16] = c11[v];
  }
}

// ---------------------------------------------------------------------------
// Kernel 4: symmetrize + row-normalize + emit edge list / batch vector.
// One wave per (graph, row k); lane = column l; wave32 shfl_xor reduction.
// ---------------------------------------------------------------------------
__global__ void k_finalize(const float* __restrict__ araw,
                           float* __restrict__ eas,
                           int* __restrict__ rows, int* __restrict__ cols,
                           int* __restrict__ bsup) {
  const int widx = (blockIdx.x * 256 + threadIdx.x) >> 5;
  const int l = threadIdx.x & 31;
  const int g = widx >> 5, k = widx & 31;
  const float* a = araw + (size_t)g * Kc * Kc;

  float v = 0.5f * (a[k * Kc + l] + a[l * Kc + k]);
  float sum = v;
#pragma unroll
  for (int off = 16; off > 0; off >>= 1) sum += __shfl_xor(sum, off);
  const float out = v / fmaxf(sum, EPSV);

  const size_t idx = (size_t)g * Kc * Kc + (size_t)k * Kc + l;
  eas[idx]  = out;
  rows[idx] = g * Kc + k;
  cols[idx] = g * Kc + l;
  if (l == 0) bsup[g * Kc + k] = g;
}

// ---------------------------------------------------------------------------
extern "C" void kernel_launch(void* const* d_in, const int* in_sizes, int n_in,
                              void* d_out, int out_size, void* d_ws, size_t ws_size,
                              hipStream_t stream) {
  const float* x  = (const float*)d_in[0];
  const float* ea = (const float*)d_in[1];
  const float* W  = (const float*)d_in[2];
  const float* bi = (const float*)d_in[3];
  const int*   ei = (const int*)d_in[4];
  const int* src = ei;
  const int* dst = ei + Ee;

  // outputs, concatenated flat in return order
  float* out  = (float*)d_out;
  float* xs   = out;                                 // [B*K, C]   = 1048576 f32
  int*   rows = (int*)(out + 1048576);               // [B*K*K]    =  262144 i32
  int*   cols = (int*)(out + 1048576 + 262144);      // [B*K*K]    =  262144 i32
  float* eas  = out + 1572864;                       // [B*K*K]    =  262144 f32
  int*   bsup = (int*)(out + 1835008);               // [B*K]      =    8192 i32
  float* s    = out + 1843200;                       // [N, K]     = 4194304 f32

  // workspace: t = s^T A  [B, K, PER] then adj_raw [B, K, K]  (~17 MB total)
  float* t    = (float*)d_ws;
  float* araw = t + (size_t)Bg * Kc * PER;

  const int tN4 = (Bg * Kc * PER) / 4;
  k_zero<<<(tN4 + 255) / 256, 256, 0, stream>>>((float4*)t, tN4);
  k_logits_softmax<<<Nn / 128, 256, 0, stream>>>(x, W, bi, s);
  k_scatter_t<<<Ee * 32 / 256, 256, 0, stream>>>(src, dst, ea, s, t);
  k_xsuper<<<Bg, 256, 0, stream>>>(x, s, xs);
  k_adjsuper<<<Bg * 32 / 256, 256, 0, stream>>>(t, s, araw);
  k_finalize<<<Bg * Kc * 32 / 256, 256, 0, stream>>>(araw, eas, rows, cols, bsup);
}